// Attention_2439541424217
// MI455X (gfx1250) — compile-verified
//
#include <hip/hip_runtime.h>

typedef __attribute__((ext_vector_type(2))) float v2f;
typedef __attribute__((ext_vector_type(8))) float v8f;

#define NB 64
#define TT 1024
#define DV 512
#define DA 128
#define DQ 1024
#define NF 32
#define KW 31
#define WIN 33           // 2*WINDOW_RANGE + 1
#define MROWS 48         // window rows padded to 3 x 16 WMMA tiles
#define ASTRIDE 516      // 512 + 4 floats pad -> conflict-free strided LDS reads

__global__ __launch_bounds__(256) void attn_win_wmma_kernel(
    const float* __restrict__ query,
    const float* __restrict__ attw,
    const float* __restrict__ current_pos,
    const float* __restrict__ value,
    const float* __restrict__ key_w,
    const float* __restrict__ key_b,
    const float* __restrict__ query_w,
    const float* __restrict__ query_b,
    const float* __restrict__ conv_w,
    const float* __restrict__ dense_w,
    const float* __restrict__ dense_b,
    const float* __restrict__ v_w,
    const int*   __restrict__ value_len,
    float* __restrict__ out_ctx,     // [B,512]
    float* __restrict__ out_align,   // [B,1024]
    float* __restrict__ out_cat,     // [B,1024,2]
    float* __restrict__ out_pos)     // [B]
{
  extern __shared__ __align__(16) float s_val[];   // MROWS x ASTRIDE (dynamic)

  __shared__ float s_query[DQ];
  __shared__ float s_convw[KW * 2 * NF];
  __shared__ float s_densew[NF * DA];
  __shared__ float s_vw[DA];
  __shared__ float s_aw[64 * 2];                   // 63 used (window +-15 halo)
  __shared__ float s_loc[WIN * NF];
  __shared__ float s_red[256];
  __shared__ float s_qkb[DA];
  __shared__ float s_procadd[WIN * DA];
  __shared__ float s_energy[MROWS];
  __shared__ float s_align[MROWS];
  __shared__ float s_scal[2];                      // max, 1/sum

  const int b   = blockIdx.x;
  const int tid = threadIdx.x;

  // ---- dynamic window (matches reference: clip, round-half-even) ----
  const int len = value_len[b];
  float cp = current_pos[b];
  cp = fminf(fmaxf(cp, 16.0f), (float)(len - 17));
  const int start = (int)rintf(fmaxf(cp - 16.0f, 0.0f));

  // ---- stage phase ----
  // Value window -> LDS via CDNA5 async copy (ASYNCcnt, no VGPR round-trip).
  // Overlaps with weight staging + query projection + conv below; drained with
  // s_wait_asynccnt right before the WMMA phase barrier.
  const float* vbase = value + ((size_t)b * TT) * DV;
  for (int i = tid; i < (MROWS * DV) / 4; i += 256) {
    int row = i >> 7;                 // 128 float4 per row
    int c4  = (i & 127) << 2;
    if (row < WIN) {
      unsigned lds_addr = (unsigned)(unsigned long long)(const void*)&s_val[row * ASTRIDE + c4];
      unsigned goff     = (unsigned)(((start + row) * DV + c4) * 4);
      asm volatile("global_load_async_to_lds_b128 %0, %1, %2"
                   :: "v"(lds_addr), "v"(goff), "s"(vbase) : "memory");
    } else {
      // pad rows feed only discarded C rows; zero for determinism
      float4 z; z.x = 0.f; z.y = 0.f; z.z = 0.f; z.w = 0.f;
      *(float4*)&s_val[row * ASTRIDE + c4] = z;
    }
  }
  for (int i = tid; i < DQ; i += 256)          s_query[i]  = query[(size_t)b * DQ + i];
  for (int i = tid; i < KW * 2 * NF; i += 256) s_convw[i]  = conv_w[i];
  for (int i = tid; i < NF * DA; i += 256)     s_densew[i] = dense_w[i];
  if (tid < DA) s_vw[tid] = v_w[tid];
  if (tid < 126) {
    int p = start - 15 + (tid >> 1);
    s_aw[tid] = (p >= 0 && p < TT) ? attw[((size_t)b * TT + p) * 2 + (tid & 1)] : 0.f;
  }
  if (tid < MROWS) s_energy[tid] = 0.f;
  __syncthreads();

  // ---- query projection partials (2 threads/column) + location conv ----
  {
    int d = tid & 127, half = tid >> 7;
    const float* qw = query_w + (size_t)(half * 512) * DA + d;
    const float* qv = s_query + half * 512;
    float acc = 0.f;
    for (int i = 0; i < 512; ++i) acc += qv[i] * qw[(size_t)i * DA];
    s_red[tid] = acc;
  }
  for (int i = tid; i < WIN * NF; i += 256) {
    int rt = i >> 5, f = i & 31;
    float acc = 0.f;
    for (int k = 0; k < KW; ++k) {
      acc += s_aw[(rt + k) * 2 + 0] * s_convw[(k * 2 + 0) * NF + f];
      acc += s_aw[(rt + k) * 2 + 1] * s_convw[(k * 2 + 1) * NF + f];
    }
    s_loc[i] = fmaxf(acc, 0.f);       // relu
  }
  __syncthreads();
  if (tid < DA)
    s_qkb[tid] = s_red[tid] + s_red[tid + 128] + query_b[tid] + key_b[tid] + dense_b[tid];
  __syncthreads();

  // processed minus key, per (t,d): loc @ dense_w + all biases + q
  for (int i = tid; i < WIN * DA; i += 256) {
    int t = i >> 7, d = i & 127;
    float acc = s_qkb[d];
    const float* lw = s_densew + d;
    const float* lr = s_loc + t * NF;
    for (int f = 0; f < NF; ++f) acc += lr[f] * lw[f * DA];
    s_procadd[i] = acc;
  }

  // drain the async value-window copy, then make it visible to all waves
#if __has_builtin(__builtin_amdgcn_s_wait_asynccnt)
  __builtin_amdgcn_s_wait_asynccnt(0);
#else
  asm volatile("s_wait_asynccnt 0x0" ::: "memory");
#endif
  __syncthreads();

  // ---- key GEMM [48x512]x[512x128] via V_WMMA_F32_16X16X4_F32, fused energy ----
  {
    const int wave = tid >> 5;        // 8 waves -> 8 N tiles
    const int lane = tid & 31;
    const int h    = lane >> 4;       // K-pair selector per ISA A/B layout
    const int l15  = lane & 15;
    const int col  = wave * 16 + l15; // absolute d
    v8f acc0 = {}, acc1 = {}, acc2 = {};
    const float* kwp = key_w + col;
    const float* a0 = s_val + (0 * 16 + l15) * ASTRIDE;
    const float* a1 = s_val + (1 * 16 + l15) * ASTRIDE;
    const float* a2 = s_val + (2 * 16 + l15) * ASTRIDE;
#pragma unroll 4
    for (int kk = 0; kk < DV; kk += 4) {
      const int k0 = kk + 2 * h;
      v2f bvec; bvec.x = kwp[(size_t)k0 * DA]; bvec.y = kwp[(size_t)(k0 + 1) * DA];
      v2f av;
      av.x = a0[k0]; av.y = a0[k0 + 1];
      acc0 = __builtin_amdgcn_wmma_f32_16x16x4_f32(false, av, false, bvec, (short)0, acc0, false, false);
      av.x = a1[k0]; av.y = a1[k0 + 1];
      acc1 = __builtin_amdgcn_wmma_f32_16x16x4_f32(false, av, false, bvec, (short)0, acc1, false, false);
      av.x = a2[k0]; av.y = a2[k0 + 1];
      acc2 = __builtin_amdgcn_wmma_f32_16x16x4_f32(false, av, false, bvec, (short)0, acc2, false, false);
    }
    // epilogue: energy[t] += tanh(key + rest) * v_w[d], reduce over d via LDS atomics
#pragma unroll
    for (int r = 0; r < 8; ++r) {
      const int mbase = r + 8 * h;    // C/D layout: VGPR r holds rows r and r+8
      {
        int m = mbase;
        atomicAdd(&s_energy[m], tanhf(acc0[r] + s_procadd[m * DA + col]) * s_vw[col]);
      }
      {
        int m = 16 + mbase;
        atomicAdd(&s_energy[m], tanhf(acc1[r] + s_procadd[m * DA + col]) * s_vw[col]);
      }
      {
        int m = 32 + mbase;
        if (m < WIN)
          atomicAdd(&s_energy[m], tanhf(acc2[r] + s_procadd[m * DA + col]) * s_vw[col]);
      }
    }
  }
  __syncthreads();

  // ---- softmax over the 33 in-window energies ----
  if (tid == 0) {
    float mx = -3.4e38f;
    for (int t = 0; t < WIN; ++t) mx = fmaxf(mx, s_energy[t]);
    s_scal[0] = mx;
  }
  __syncthreads();
  if (tid < WIN) s_align[tid] = expf(s_energy[tid] - s_scal[0]);
  __syncthreads();
  if (tid == 0) {
    float sm = 0.f;
    for (int t = 0; t < WIN; ++t) sm += s_align[t];
    s_scal[1] = 1.0f / sm;
  }
  __syncthreads();
  if (tid < WIN) s_align[tid] *= s_scal[1];
  __syncthreads();

  // ---- outputs ----
  // context[b,c] = sum_t align[t] * value[b, start+t, c]
  for (int c = tid; c < DV; c += 256) {
    float acc = 0.f;
    for (int t = 0; t < WIN; ++t) acc += s_align[t] * s_val[t * ASTRIDE + c];
    out_ctx[(size_t)b * DV + c] = acc;
  }
  // align is exactly zero outside the window; new_cat ch0=align, ch1=aw1+align
  for (int t = tid; t < TT; t += 256) {
    float a = (t >= start && t < start + WIN) ? s_align[t - start] : 0.f;
    float aw1 = attw[((size_t)b * TT + t) * 2 + 1];
    out_align[(size_t)b * TT + t]         = a;
    out_cat[((size_t)b * TT + t) * 2 + 0] = a;
    out_cat[((size_t)b * TT + t) * 2 + 1] = aw1 + a;
  }
  if (tid == 0) {
    float np = 0.f;
    for (int t = 0; t < WIN; ++t) np += s_align[t] * (float)(start + t);
    out_pos[b] = np;
  }
}

extern "C" void kernel_launch(void* const* d_in, const int* in_sizes, int n_in,
                              void* d_out, int out_size, void* d_ws, size_t ws_size,
                              hipStream_t stream) {
  const float* query       = (const float*)d_in[0];
  const float* attw        = (const float*)d_in[1];
  const float* current_pos = (const float*)d_in[2];
  const float* value       = (const float*)d_in[3];
  const float* key_w       = (const float*)d_in[4];
  const float* key_b       = (const float*)d_in[5];
  const float* query_w     = (const float*)d_in[6];
  const float* query_b     = (const float*)d_in[7];
  const float* conv_w      = (const float*)d_in[8];
  const float* dense_w     = (const float*)d_in[9];
  const float* dense_b     = (const float*)d_in[10];
  const float* v_w         = (const float*)d_in[11];
  const int*   value_len   = (const int*)d_in[12];

  float* out       = (float*)d_out;          // outputs flat in return order
  float* out_ctx   = out;                    // [64,512]
  float* out_align = out_ctx + NB * DV;      // [64,1024,1]
  float* out_cat   = out_align + NB * TT;    // [64,1024,2]
  float* out_pos   = out_cat + NB * TT * 2;  // [64,1,1]

  size_t shmem = (size_t)MROWS * ASTRIDE * sizeof(float);  // 99 KB value window
  attn_win_wmma_kernel<<<dim3(NB), dim3(256), shmem, stream>>>(
      query, attw, current_pos, value, key_w, key_b, query_w, query_b,
      conv_w, dense_w, dense_b, v_w, value_len,
      out_ctx, out_align, out_cat, out_pos);
}